// CoreGCN_87840671138056
// MI455X (gfx1250) — compile-verified
//
#include <hip/hip_runtime.h>

// CDNA5 (gfx1250, wave32) 2-layer GCN.
// Pipeline: degree atomics -> edge norm -> [memset agg; wave-per-edge scatter
// (GLOBAL_ATOMIC_ADD_F32 into L2-resident agg); WMMA f32 16x16x4 GEMM] x2.
// All dense math uses V_WMMA_F32_16X16X4_F32 to keep full f32 precision.
// W is staged in LDS in paired-K layout (float2 = {W[k][c], W[k+1][c]}) with a
// bank-parity swizzle so each B fragment is a single conflict-free ds_load_b64
// directly into the WMMA source register pair.

typedef __attribute__((ext_vector_type(2))) float v2f;
typedef __attribute__((ext_vector_type(8))) float v8f;

#define N_FEAT 128

__device__ __forceinline__ float selu_f(float x) {
    const float lam = 1.0507009873554804934193349852946f;
    const float alp = 1.6732632423543772848170429916717f;
    return x > 0.0f ? lam * x : lam * alp * (expf(x) - 1.0f);
}

__device__ __forceinline__ void atomic_add_f32(float* p, float v) {
    __hip_atomic_fetch_add(p, v, __ATOMIC_RELAXED, __HIP_MEMORY_SCOPE_AGENT);
}

// ---------- weighted + unweighted degree accumulation (one thread / edge) ----------
__global__ void degrees_kernel(const int* __restrict__ src, const int* __restrict__ dst,
                               const float* __restrict__ w,
                               float* __restrict__ wdeg_out, float* __restrict__ wdeg_in,
                               float* __restrict__ out_deg, float* __restrict__ in_deg,
                               int nedges) {
    int e = blockIdx.x * blockDim.x + threadIdx.x;
    if (e >= nedges) return;
    int s = src[e], d = dst[e];
    float we = w[e];
    atomic_add_f32(wdeg_out + s, we);
    atomic_add_f32(wdeg_in + d, we);
    atomic_add_f32(out_deg + s, 1.0f);
    atomic_add_f32(in_deg + d, 1.0f);
}

// ---------- EdgeWeightNorm(norm='both') ----------
__global__ void edge_norm_kernel(const int* __restrict__ src, const int* __restrict__ dst,
                                 const float* __restrict__ w,
                                 const float* __restrict__ wdeg_out,
                                 const float* __restrict__ wdeg_in,
                                 float* __restrict__ nw, int nedges) {
    int e = blockIdx.x * blockDim.x + threadIdx.x;
    if (e >= nedges) return;
    const float EPS = 1e-12f;
    int s = src[e], d = dst[e];
    nw[e] = w[e] * rsqrtf(fmaxf(wdeg_out[s], EPS)) * rsqrtf(fmaxf(wdeg_in[d], EPS));
}

// ---------- SpMM scatter: one wave (32 lanes) per edge, float4 gather, 4 atomics/lane ----------
template <bool SELU_IN>
__global__ void scatter_kernel(const float* __restrict__ feat,
                               const int* __restrict__ src, const int* __restrict__ dst,
                               const float* __restrict__ nw, const float* __restrict__ out_deg,
                               float* __restrict__ agg, int nedges) {
    long long gtid = (long long)blockIdx.x * blockDim.x + threadIdx.x;
    int e = (int)(gtid >> 5);
    if (e >= nedges) return;
    int lane = threadIdx.x & 31;
    int s = src[e], d = dst[e];
    // fuse rsqrt(out_deg[src]) into the per-edge coefficient
    float coef = nw[e] * rsqrtf(fmaxf(out_deg[s], 1.0f));
    float4 v4 = ((const float4*)(feat + (long long)s * N_FEAT))[lane];
    float vals[4] = {v4.x, v4.y, v4.z, v4.w};
    float* base = agg + (long long)d * N_FEAT + lane * 4;
#pragma unroll
    for (int j = 0; j < 4; ++j) {
        float v = vals[j];
        if (SELU_IN) v = selu_f(v);
        atomic_add_f32(base + j, coef * v);
    }
}

// ---------- WMMA GEMM: out = selu?(diag(rsqrt(in_deg)) * agg @ W + b) ----------
// 256 threads = 8 wave32; each wave computes a 16x128 output strip via 8 col-tiles
// of V_WMMA_F32_16X16X4_F32 over K=128. W staged in 64KB LDS, paired-K + swizzle.
template <bool SELU_OUT>
__global__ void __launch_bounds__(256)
gemm_wmma_kernel(const float* __restrict__ A, const float* __restrict__ in_deg,
                 const float* __restrict__ W, const float* __restrict__ bias,
                 float* __restrict__ out, int n) {
    // Wp[kp][col'] = {W[2kp][col], W[2kp+1][col]}, col' = col ^ ((kp&1)<<4).
    // Lanes 0-15 always read even kp (banks 0-31), lanes 16-31 odd kp; the
    // bit-4 swizzle moves odd-kp accesses to banks 32-63 -> conflict-free b64.
    __shared__ __align__(16) float Wlds[N_FEAT * N_FEAT];  // 64 KB
    float2* Wp = (float2*)Wlds;                            // [64][128] pairs

    int tid = threadIdx.x;
    for (int i = tid; i < (N_FEAT / 2) * N_FEAT; i += 256) {
        int kp = i >> 7, col = i & 127;
        float2 p;
        p.x = W[(2 * kp) * N_FEAT + col];
        p.y = W[(2 * kp + 1) * N_FEAT + col];
        Wp[i ^ ((kp & 1) << 4)] = p;                       // apply bank swizzle
    }
    __syncthreads();

    int wave  = tid >> 5;
    int lane  = tid & 31;
    int lhalf = lane >> 4;    // 0: K=k..k+1 / rows 0..7 of D ; 1: K=k+2..k+3 / rows 8..15
    int lrow  = lane & 15;
    int rowBase = blockIdx.x * 128 + wave * 16;
    int row  = rowBase + lrow;
    int rowc = row < n ? row : n - 1;           // clamp: keep EXEC all-ones for WMMA
    float scale = rsqrtf(fmaxf(in_deg[rowc], 1.0f));
    const float* Arow = A + (long long)rowc * N_FEAT;

    // swizzle-resolved base pointers: even col-tiles use +swz, odd use -swz
    int swz = lhalf << 4;     // 16 pairs for the odd-kp half-wave
    v8f acc[8] = {};

    for (int k = 0; k < N_FEAT; k += 4) {
        int kp = (k >> 1) + lhalf;
        // A fragment (16x4): lane holds A[row][kk], A[row][kk+1], in-deg norm fused
        v2f a;
        a.x = Arow[k + 2 * lhalf] * scale;
        a.y = Arow[k + 2 * lhalf + 1] * scale;
        const float2* Brow = Wp + (kp << 7);
        const float2* p0 = Brow + lrow + swz;   // even t: (t*16+lrow) ^ swz = t*16+lrow+swz
        const float2* p1 = Brow + lrow - swz;   // odd  t: (t*16+lrow) ^ swz = t*16+lrow-swz
#pragma unroll
        for (int t = 0; t < 8; ++t) {
            float2 bp = ((t & 1) ? p1 : p0)[t * 16];   // ds_load_b64, imm offset t*128B
            v2f b;
            b.x = bp.x;
            b.y = bp.y;
            acc[t] = __builtin_amdgcn_wmma_f32_16x16x4_f32(
                false, a, false, b, (short)0, acc[t], false, false);
        }
    }

    // Epilogue: D layout -> VGPR i is row (rowBase + lhalf*8 + i), col = t*16 + lrow
#pragma unroll
    for (int t = 0; t < 8; ++t) {
        int col = t * 16 + lrow;
        float bv = bias[col];
#pragma unroll
        for (int i = 0; i < 8; ++i) {
            int r = rowBase + lhalf * 8 + i;
            if (r < n) {
                float v = acc[t][i] + bv;
                if (SELU_OUT) v = selu_f(v);
                out[(long long)r * N_FEAT + col] = v;
            }
        }
    }
}

extern "C" void kernel_launch(void* const* d_in, const int* in_sizes, int n_in,
                              void* d_out, int out_size, void* d_ws, size_t ws_size,
                              hipStream_t stream) {
    const float* x   = (const float*)d_in[0];
    const int*   src = (const int*)d_in[1];
    const int*   dst = (const int*)d_in[2];
    const float* ew  = (const float*)d_in[3];
    const float* W1  = (const float*)d_in[4];
    const float* b1  = (const float*)d_in[5];
    const float* W2  = (const float*)d_in[6];
    const float* b2  = (const float*)d_in[7];
    float* out = (float*)d_out;

    int n = in_sizes[0] / N_FEAT;   // 50000
    int e = in_sizes[1];            // 800000

    // workspace layout (floats): 4 degree arrays | nw[E] | agg[N*128] | h1[N*128]
    float* ws       = (float*)d_ws;
    float* wdeg_out = ws;
    float* wdeg_in  = ws + (long long)n;
    float* out_deg  = ws + 2LL * n;
    float* in_deg   = ws + 3LL * n;
    float* nw       = ws + 4LL * n;
    float* agg      = nw + (long long)e;
    float* h1       = agg + (long long)n * N_FEAT;

    const int tpb = 256;
    int eblocks = (e + tpb - 1) / tpb;
    long long sthreads = (long long)e * 32;                   // one wave per edge
    int sblocks = (int)((sthreads + tpb - 1) / tpb);
    int gblocks = (n + 127) / 128;

    // degrees + edge-weight normalization
    hipMemsetAsync(wdeg_out, 0, sizeof(float) * 4 * (size_t)n, stream);
    degrees_kernel<<<eblocks, tpb, 0, stream>>>(src, dst, ew, wdeg_out, wdeg_in,
                                                out_deg, in_deg, e);
    edge_norm_kernel<<<eblocks, tpb, 0, stream>>>(src, dst, ew, wdeg_out, wdeg_in, nw, e);

    // ---- layer 1: h1 = agg(x) @ W1 + b1 (no activation stored) ----
    hipMemsetAsync(agg, 0, sizeof(float) * (size_t)n * N_FEAT, stream);
    scatter_kernel<false><<<sblocks, tpb, 0, stream>>>(x, src, dst, nw, out_deg, agg, e);
    gemm_wmma_kernel<false><<<gblocks, 256, 0, stream>>>(agg, in_deg, W1, b1, h1, n);

    // ---- layer 2: out = selu( agg(selu(h1)) @ W2 + b2 ) ----
    hipMemsetAsync(agg, 0, sizeof(float) * (size_t)n * N_FEAT, stream);
    scatter_kernel<true><<<sblocks, tpb, 0, stream>>>(h1, src, dst, nw, out_deg, agg, e);
    gemm_wmma_kernel<true><<<gblocks, 256, 0, stream>>>(agg, in_deg, W2, b2, out, n);
}